// GraphConvolution_23235773071980
// MI455X (gfx1250) — compile-verified
//
#include <hip/hip_runtime.h>

#define N_NODES 8192
#define D 256
#define BATCH 8
#define MAXDEG 256
#define WORDS_PER_ROW (N_NODES / 64)

typedef float v2f __attribute__((ext_vector_type(2)));
typedef float v8f __attribute__((ext_vector_type(8)));

// ---------------- bitmask zero ----------------
__global__ __launch_bounds__(256) void zero_mask_kernel(unsigned long long* __restrict__ mask) {
    size_t i = (size_t)blockIdx.x * blockDim.x + threadIdx.x;
    mask[i] = 0ull;
}

// ---------------- edge scatter (set semantics via atomicOr -> dedupe) ----------------
__global__ __launch_bounds__(256) void scatter_kernel(const long long* __restrict__ edges,
                                                      unsigned long long* __restrict__ mask,
                                                      int nE) {
    int i = blockIdx.x * blockDim.x + threadIdx.x;
    if (i >= nE) return;
    long long s = edges[2 * i + 0];
    long long d = edges[2 * i + 1];
    size_t bit = (size_t)s * N_NODES + (size_t)d;
    atomicOr(&mask[bit >> 6], 1ull << (bit & 63));
}

// ---------------- deterministic per-row neighbor lists (ascending m) ----------------
__global__ __launch_bounds__(256) void rowlist_kernel(const unsigned long long* __restrict__ mask,
                                                      int* __restrict__ rowlist,
                                                      int* __restrict__ rowcnt) {
    int n = blockIdx.x * blockDim.x + threadIdx.x;
    if (n >= N_NODES) return;
    const unsigned long long* row = mask + (size_t)n * WORDS_PER_ROW;
    int* lst = rowlist + (size_t)n * MAXDEG;
    int cnt = 0;
    for (int w = 0; w < WORDS_PER_ROW; ++w) {
        unsigned long long bits = row[w];
        while (bits) {
            int b = __ffsll(bits) - 1;
            bits &= bits - 1;
            if (cnt < MAXDEG) lst[cnt] = (w << 6) + b;
            ++cnt;
        }
    }
    rowcnt[n] = cnt < MAXDEG ? cnt : MAXDEG;
}

// ---------------- xw = x @ W via V_WMMA_F32_16X16X4_F32 ----------------
// one wave: 16 rows x 64 cols; block = 4 waves; grid = (65536/16/4, 256/64)
__global__ __launch_bounds__(128) void gemm_kernel(const float* __restrict__ x,
                                                   const float* __restrict__ W,
                                                   float* __restrict__ xw) {
    const int wave = threadIdx.x >> 5;
    const int lane = threadIdx.x & 31;
    const int half = lane >> 4;   // 0: K pair {0,1}, 1: K pair {2,3}
    const int r    = lane & 15;   // row (A) / col (B)

    const size_t row0 = ((size_t)blockIdx.x * 4 + wave) * 16;  // batch folded into rows
    const int    col0 = blockIdx.y * 64;

    const float* xrow = x + (row0 + r) * D;

    v8f acc[4] = {};

    for (int k0 = 0; k0 < D; k0 += 4) {
        v2f a;
        a.x = xrow[k0 + 2 * half + 0];
        a.y = xrow[k0 + 2 * half + 1];
#pragma unroll
        for (int t = 0; t < 4; ++t) {
            const int c = col0 + t * 16 + r;
            v2f bb;
            bb.x = W[(k0 + 2 * half + 0) * D + c];
            bb.y = W[(k0 + 2 * half + 1) * D + c];
            acc[t] = __builtin_amdgcn_wmma_f32_16x16x4_f32(
                false, a, false, bb, (short)0, acc[t], false, false);
        }
    }

#pragma unroll
    for (int t = 0; t < 4; ++t) {
        const int c = col0 + t * 16 + r;
#pragma unroll
        for (int v = 0; v < 8; ++v) {
            xw[(row0 + v + 8 * half) * D + c] = acc[t][v];
        }
    }
}

// ---------------- sparse aggregate + LeakyReLU + LayerNorm ----------------
// one block per (b, n): thread d owns channel d. Gathers hit L2 (xw = 64MB < 192MB L2).
__global__ __launch_bounds__(256) void agg_kernel(const float* __restrict__ xw,
                                                  const int* __restrict__ rowlist,
                                                  const int* __restrict__ rowcnt,
                                                  const float* __restrict__ gamma,
                                                  const float* __restrict__ beta,
                                                  float* __restrict__ out) {
    const int n = blockIdx.x & (N_NODES - 1);
    const int b = blockIdx.x >> 13;
    const int d = threadIdx.x;

    const int cnt = rowcnt[n];
    const int* lst = rowlist + (size_t)n * MAXDEG;
    const float* xwb = xw + (size_t)b * N_NODES * D;

    float acc = 0.f;
    for (int i = 0; i < cnt; ++i) {
        acc += xwb[(size_t)lst[i] * D + d];
    }

    // LeakyReLU(0.1)
    float h = acc >= 0.f ? acc : 0.1f * acc;

    // LayerNorm over 256 channels: wave32 shuffle reduce, then 8 partials in LDS
    float s = h, q = h * h;
    for (int off = 16; off; off >>= 1) {
        s += __shfl_down(s, off, 32);
        q += __shfl_down(q, off, 32);
    }
    __shared__ float s1[8], s2[8];
    if ((threadIdx.x & 31) == 0) {
        s1[threadIdx.x >> 5] = s;
        s2[threadIdx.x >> 5] = q;
    }
    __syncthreads();
    float sum = 0.f, sq = 0.f;
#pragma unroll
    for (int i = 0; i < 8; ++i) { sum += s1[i]; sq += s2[i]; }

    const float mean = sum * (1.f / D);
    const float var  = sq * (1.f / D) - mean * mean;       // population var (jnp.var)
    const float inv  = __frsqrt_rn(var + 1e-5f);

    out[(size_t)blockIdx.x * D + d] = gamma[d] * (h - mean) * inv + beta[d];
}

extern "C" void kernel_launch(void* const* d_in, const int* in_sizes, int n_in,
                              void* d_out, int out_size, void* d_ws, size_t ws_size,
                              hipStream_t stream) {
    const float*     x     = (const float*)d_in[0];
    const float*     W     = (const float*)d_in[1];
    const float*     gamma = (const float*)d_in[2];
    const float*     beta  = (const float*)d_in[3];
    const long long* edges = (const long long*)d_in[4];  // int64 (src,dst) pairs
    const int nE = in_sizes[4] / 2;

    char* ws = (char*)d_ws;
    float*              xw      = (float*)ws;                          // 64 MiB
    unsigned long long* mask    = (unsigned long long*)(ws + (64ull << 20)); // 8 MiB
    int*                rowlist = (int*)(ws + (72ull << 20));          // 8 MiB
    int*                rowcnt  = (int*)(ws + (80ull << 20));          // 32 KiB

    // 1) clear adjacency bitmask (re-done every call: no cross-call state)
    zero_mask_kernel<<<(N_NODES * WORDS_PER_ROW) / 256, 256, 0, stream>>>(mask);

    // 2) scatter edges (idempotent atomicOr -> deterministic, dedupes duplicates)
    scatter_kernel<<<(nE + 255) / 256, 256, 0, stream>>>(edges, mask, nE);

    // 3) compact each row into an ascending neighbor list (deterministic sum order)
    rowlist_kernel<<<N_NODES / 256, 256, 0, stream>>>(mask, rowlist, rowcnt);

    // 4) xw = x @ W  (f32 WMMA); rows = b*N = 65536
    dim3 ggrid((BATCH * N_NODES / 16) / 4, D / 64);
    gemm_kernel<<<ggrid, 128, 0, stream>>>(x, W, xw);

    // 5) sparse aggregate + LeakyReLU + LayerNorm
    agg_kernel<<<BATCH * N_NODES, 256, 0, stream>>>(xw, rowlist, rowcnt, gamma, beta,
                                                    (float*)d_out);
}